// SwinTransformer_25598005084367
// MI455X (gfx1250) — compile-verified
//
#include <hip/hip_runtime.h>

// ---------------------------------------------------------------------------
// SwinUNETR-like reference, MI455X (gfx1250, wave32).
//
// Front (patch-embed + window attention) collapses exactly to a constant
// per-window row  S[t] = ln2_b[0] * colsum(out_w)[t]  because the inner LN is
// over a size-1 axis.  Remaining work: conv/deconv U-Net tail (~2.3 GFLOP
// fp32, ~35 MB intermediates -> fully L2-resident, latency bound).
// Convs run as implicit-GEMM on V_WMMA_F32_16X16X4_F32 (exact fp32 path);
// transposed convs are the same kernel with implicit lhs-dilation (parity
// gather) and pad_lo=2, skip-adds folded into the gather.
//
// Round-2: per-tap address hoisting + branchless gathers; Cin templated so the
// cin loop is division-free and fully unrolled (4 WMMAs per 4-wide K step).
// ---------------------------------------------------------------------------

typedef float v2f __attribute__((ext_vector_type(2)));
typedef float v8f __attribute__((ext_vector_type(8)));

#define EPS_LN 1e-6f

// ---------------------------------------------------------------------------
// Fast implicit-GEMM conv, Cin % 4 == 0.  One wave per block computes a
// 16-channel x 64-voxel C tile (4 WMMA N-tiles).  WMMA register mapping:
//   A (16x4, MxK): m = lane&15, k = (lane>>4)*2 + r
//   B (4x16, KxN): n = lane&15, k = (lane>>4)*2 + r
//   C/D:           n = lane&15, m = (lane>>4)*8 + j
// ---------------------------------------------------------------------------
template <int CIN>
__global__ void __launch_bounds__(32)
conv3d_wmma_fast(const float* __restrict__ in, const float* __restrict__ in2,
                 const float* __restrict__ wgt, const float* __restrict__ bias,
                 float* __restrict__ out,
                 int inD, int inH, int inW,
                 int outD, int outH, int outW, int Cout,
                 int stride, int padlo, int dilate) {
  const int lane = threadIdx.x & 31;
  const int m = lane & 15;
  const int khalf = lane >> 4;
  const int chanBase = blockIdx.y * 16;
  const int chan = chanBase + m;
  const bool chanOK = chan < Cout;
  const int chanC = chanOK ? chan : 0;  // clamped -> always-safe weight loads
  const int voxBase = blockIdx.x * 64;
  const int numVox = outD * outH * outW;

  // Branchless skip-add: in2p always dereferenceable, scaled by 0 if unused.
  const float* __restrict__ in2p = in2 ? in2 : in;
  const float skipScale = in2 ? 1.0f : 0.0f;

  // Per-tile voxel coordinates, computed once.
  int od[4], oh[4], ow[4];
  bool vOK[4];
#pragma unroll
  for (int t = 0; t < 4; ++t) {
    const int vox = voxBase + t * 16 + m;
    vOK[t] = vox < numVox;
    const int vv = vOK[t] ? vox : 0;
    od[t] = vv / (outH * outW);
    const int r = vv - od[t] * outH * outW;
    oh[t] = r / outW;
    ow[t] = r - oh[t] * outW;
  }

  const int limD = dilate ? (inD << 1) : inD;
  const int limH = dilate ? (inH << 1) : inH;
  const int limW = dilate ? (inW << 1) : inW;

  v8f acc0 = v8f{}, acc1 = v8f{}, acc2 = v8f{}, acc3 = v8f{};

  for (int tap = 0; tap < 27; ++tap) {
    const int kd = tap / 9;
    const int t9 = tap - kd * 9;
    const int kh = t9 / 3;
    const int kw = t9 - kh * 3;

    // Per-tile input base element index (clamped to 0 when invalid).
    int ibase[4];
    float isel[4];
#pragma unroll
    for (int t = 0; t < 4; ++t) {
      int sd = od[t] * stride + kd - padlo;
      int sh = oh[t] * stride + kh - padlo;
      int sw = ow[t] * stride + kw - padlo;
      bool ok = vOK[t] && sd >= 0 && sd < limD && sh >= 0 && sh < limH &&
                sw >= 0 && sw < limW;
      if (dilate) {
        ok = ok && (((sd | sh | sw) & 1) == 0);  // zero-stuffed positions
        sd >>= 1; sh >>= 1; sw >>= 1;
      }
      ibase[t] = ok ? ((sd * inH + sh) * inW + sw) * CIN : 0;
      isel[t] = ok ? 1.0f : 0.0f;
    }

    const float* __restrict__ wrow = wgt + (size_t)(tap * CIN) * Cout + chanC;

#pragma unroll
    for (int c0 = 0; c0 < CIN; c0 += 4) {
      const int cc = c0 + khalf * 2;  // this lane's two K slots
      v2f a;
      a.x = chanOK ? wrow[(size_t)cc * Cout] : 0.0f;
      a.y = chanOK ? wrow[(size_t)(cc + 1) * Cout] : 0.0f;
      v2f b;

      b.x = isel[0] * (in[ibase[0] + cc]     + skipScale * in2p[ibase[0] + cc]);
      b.y = isel[0] * (in[ibase[0] + cc + 1] + skipScale * in2p[ibase[0] + cc + 1]);
      acc0 = __builtin_amdgcn_wmma_f32_16x16x4_f32(false, a, false, b, (short)0,
                                                   acc0, false, false);
      b.x = isel[1] * (in[ibase[1] + cc]     + skipScale * in2p[ibase[1] + cc]);
      b.y = isel[1] * (in[ibase[1] + cc + 1] + skipScale * in2p[ibase[1] + cc + 1]);
      acc1 = __builtin_amdgcn_wmma_f32_16x16x4_f32(false, a, false, b, (short)0,
                                                   acc1, false, false);
      b.x = isel[2] * (in[ibase[2] + cc]     + skipScale * in2p[ibase[2] + cc]);
      b.y = isel[2] * (in[ibase[2] + cc + 1] + skipScale * in2p[ibase[2] + cc + 1]);
      acc2 = __builtin_amdgcn_wmma_f32_16x16x4_f32(false, a, false, b, (short)0,
                                                   acc2, false, false);
      b.x = isel[3] * (in[ibase[3] + cc]     + skipScale * in2p[ibase[3] + cc]);
      b.y = isel[3] * (in[ibase[3] + cc + 1] + skipScale * in2p[ibase[3] + cc + 1]);
      acc3 = __builtin_amdgcn_wmma_f32_16x16x4_f32(false, a, false, b, (short)0,
                                                   acc3, false, false);
    }
  }

  // Epilogue: bias add + NDHWC store (guarded; divergence fine post-WMMA).
#define STORE_TILE(ACC, T)                                                     \
  {                                                                            \
    const int vox = voxBase + (T)*16 + m;                                      \
    if (vox < numVox) {                                                        \
      _Pragma("unroll") for (int j = 0; j < 8; ++j) {                          \
        const int ch = chanBase + khalf * 8 + j;                               \
        if (ch < Cout) out[vox * Cout + ch] = ACC[j] + bias[ch];               \
      }                                                                        \
    }                                                                          \
  }
  STORE_TILE(acc0, 0)
  STORE_TILE(acc1, 1)
  STORE_TILE(acc2, 2)
  STORE_TILE(acc3, 3)
#undef STORE_TILE
}

// ---------------------------------------------------------------------------
// Generic (slow) conv for Cin == 1 (first layer only, 42 MFLOP).
// ---------------------------------------------------------------------------
__device__ __forceinline__ float gather_w_g(const float* __restrict__ w, int kk,
                                            int K, int Cin, int Cout, int chan) {
  if (kk >= K || chan >= Cout) return 0.0f;
  const int tap = kk / Cin;
  const int cin = kk - tap * Cin;
  return w[(tap * Cin + cin) * Cout + chan];
}

__device__ __forceinline__ float gather_in_g(const float* __restrict__ in,
                                             int vox, int numVox, int kk, int K,
                                             int inD, int inH, int inW, int Cin,
                                             int outH, int outW,
                                             int stride, int padlo) {
  if (vox >= numVox || kk >= K) return 0.0f;
  const int tap = kk / Cin;
  const int cin = kk - tap * Cin;
  const int kd = tap / 9; const int tr = tap - kd * 9;
  const int kh = tr / 3;  const int kw = tr - kh * 3;
  const int odv = vox / (outH * outW); const int r = vox - odv * outH * outW;
  const int ohv = r / outW;            const int owv = r - ohv * outW;
  const int sd = odv * stride + kd - padlo;
  const int sh = ohv * stride + kh - padlo;
  const int sw = owv * stride + kw - padlo;
  if (sd < 0 || sd >= inD || sh < 0 || sh >= inH || sw < 0 || sw >= inW)
    return 0.0f;
  return in[((sd * inH + sh) * inW + sw) * Cin + cin];
}

__global__ void __launch_bounds__(32)
conv3d_wmma_generic(const float* __restrict__ in, const float* __restrict__ wgt,
                    const float* __restrict__ bias, float* __restrict__ out,
                    int inD, int inH, int inW, int Cin,
                    int outD, int outH, int outW, int Cout,
                    int stride, int padlo) {
  const int lane = threadIdx.x & 31;
  const int m = lane & 15;
  const int khalf = lane >> 4;
  const int chanBase = blockIdx.y * 16;
  const int voxBase = blockIdx.x * 64;
  const int numVox = outD * outH * outW;
  const int K = 27 * Cin;

  v8f acc0 = v8f{}, acc1 = v8f{}, acc2 = v8f{}, acc3 = v8f{};

  for (int k0 = 0; k0 < K; k0 += 4) {
    const int kb = k0 + khalf * 2;
    v2f a;
    a.x = gather_w_g(wgt, kb,     K, Cin, Cout, chanBase + m);
    a.y = gather_w_g(wgt, kb + 1, K, Cin, Cout, chanBase + m);
    v2f b;
#define B_TILE(ACC, T)                                                         \
    b.x = gather_in_g(in, voxBase + (T)*16 + m, numVox, kb, K, inD, inH, inW,  \
                      Cin, outH, outW, stride, padlo);                         \
    b.y = gather_in_g(in, voxBase + (T)*16 + m, numVox, kb + 1, K, inD, inH,   \
                      inW, Cin, outH, outW, stride, padlo);                    \
    ACC = __builtin_amdgcn_wmma_f32_16x16x4_f32(false, a, false, b, (short)0,  \
                                                ACC, false, false);
    B_TILE(acc0, 0)
    B_TILE(acc1, 1)
    B_TILE(acc2, 2)
    B_TILE(acc3, 3)
#undef B_TILE
  }

#define STORE_TILE(ACC, T)                                                     \
  {                                                                            \
    const int vox = voxBase + (T)*16 + m;                                      \
    if (vox < numVox) {                                                        \
      _Pragma("unroll") for (int j = 0; j < 8; ++j) {                          \
        const int ch = chanBase + khalf * 8 + j;                               \
        if (ch < Cout) out[vox * Cout + ch] = ACC[j] + bias[ch];               \
      }                                                                        \
    }                                                                          \
  }
  STORE_TILE(acc0, 0)
  STORE_TILE(acc1, 1)
  STORE_TILE(acc2, 2)
  STORE_TILE(acc3, 3)
#undef STORE_TILE
}

// S[t] = ln2_b[0] * sum_i out_w[i][t]   (exact collapse of window attention)
__global__ void front_colsum(const float* __restrict__ out_w,
                             const float* __restrict__ ln2_b,
                             float* __restrict__ S) {
  const int t = threadIdx.x;  // 64 threads
  float acc = 0.0f;
  for (int i = 0; i < 256; ++i) acc += out_w[i * 64 + t];
  S[t] = acc * ln2_b[0];
}

// x0[d,h,w] = S[(d%4)*16 + (h%4)*4 + (w%4)]
__global__ void fill_pattern(const float* __restrict__ S, float* __restrict__ x0) {
  const int idx = blockIdx.x * blockDim.x + threadIdx.x;
  if (idx >= 64 * 64 * 64) return;
  const int w = idx & 63, h = (idx >> 6) & 63, d = idx >> 12;
  x0[idx] = S[((d & 3) << 4) | ((h & 3) << 2) | (w & 3)];
}

// In-place channel LayerNorm (biased variance, eps=1e-6), one thread/voxel.
__global__ void layernorm_c(float* __restrict__ x, const float* __restrict__ s,
                            const float* __restrict__ b, int C, int numVox) {
  const int v = blockIdx.x * blockDim.x + threadIdx.x;
  if (v >= numVox) return;
  float* p = x + (size_t)v * C;
  float mean = 0.0f;
  for (int c = 0; c < C; ++c) mean += p[c];
  mean /= (float)C;
  float var = 0.0f;
  for (int c = 0; c < C; ++c) { float d2 = p[c] - mean; var += d2 * d2; }
  var /= (float)C;
  const float inv = rsqrtf(var + EPS_LN);
  for (int c = 0; c < C; ++c) p[c] = (p[c] - mean) * inv * s[c] + b[c];
}

// ---------------------------------------------------------------------------
// Host side: input binding + launch sequence.
// ---------------------------------------------------------------------------
namespace {
enum Name {
  N_X = 0, N_PE_K, N_PE_B, N_PE_LN_S, N_PE_LN_B, N_ATTN_LN_S, N_ATTN_LN_B,
  N_QKV_W, N_LN2_S, N_LN2_B, N_OUT_W,
  N_CA_K, N_CA_B, N_CB_K, N_CB_B, N_CC_K, N_CC_B, N_CD_K, N_CD_B,
  N_DA_K, N_DA_B, N_DA_LN_S, N_DA_LN_B,
  N_DB_K, N_DB_B, N_DB_LN_S, N_DB_LN_B,
  N_DC_K, N_DC_B, N_DC_LN_S, N_DC_LN_B,
  N_DD_K, N_DD_B, N_DD_LN_S, N_DD_LN_B,
  N_CO_K, N_CO_B, N_COUNT
};

const int kSz[N_COUNT] = {
  262144, 48, 48, 48, 48, 48, 48, 2359296, 1, 1, 16384,
  648, 24, 31104, 48, 124416, 96, 497664, 192,
  497664, 96, 96, 96, 124416, 48, 48, 48,
  31104, 24, 24, 24, 17496, 24, 24, 24, 648, 1
};

// params keys in lexicographic (jax pytree) order
const int kSorted[36] = {
  N_ATTN_LN_B, N_ATTN_LN_S, N_CA_B, N_CA_K, N_CB_B, N_CB_K, N_CC_B, N_CC_K,
  N_CD_B, N_CD_K, N_CO_B, N_CO_K, N_DA_B, N_DA_K, N_DA_LN_B, N_DA_LN_S,
  N_DB_B, N_DB_K, N_DB_LN_B, N_DB_LN_S, N_DC_B, N_DC_K, N_DC_LN_B, N_DC_LN_S,
  N_DD_B, N_DD_K, N_DD_LN_B, N_DD_LN_S, N_LN2_B, N_LN2_S, N_OUT_W,
  N_PE_B, N_PE_K, N_PE_LN_B, N_PE_LN_S, N_QKV_W
};
}  // namespace

extern "C" void kernel_launch(void* const* d_in, const int* in_sizes, int n_in,
                              void* d_out, int out_size, void* d_ws, size_t ws_size,
                              hipStream_t stream) {
  if (n_in < N_COUNT) return;

  // Candidate flattening orders: {x+insertion, x+sorted, sorted+x, insertion+x}
  int order[4][N_COUNT];
  for (int i = 0; i < N_COUNT; ++i) order[0][i] = i;
  order[1][0] = N_X;
  for (int i = 0; i < 36; ++i) order[1][1 + i] = kSorted[i];
  for (int i = 0; i < 36; ++i) order[2][i] = kSorted[i];
  order[2][36] = N_X;
  for (int i = 0; i < 36; ++i) order[3][i] = i + 1;
  order[3][36] = N_X;

  int chosen = 0;
  for (int c = 0; c < 4; ++c) {
    bool ok = true;
    for (int i = 0; i < N_COUNT; ++i)
      if (in_sizes[i] != kSz[order[c][i]]) { ok = false; break; }
    if (ok) { chosen = c; break; }
  }
  int pos[N_COUNT];
  for (int i = 0; i < N_COUNT; ++i) pos[order[chosen][i]] = i;
  auto P = [&](int name) -> const float* { return (const float*)d_in[pos[name]]; };

  // Workspace layout (floats): ~34.5 MB total.
  float* ws = (float*)d_ws;
  float* S  = ws;
  float* x0 = S  + 64;                   // 64^3 x 1
  float* x1 = x0 + 64 * 64 * 64;         // 32^3 x 24
  float* x2 = x1 + 32 * 32 * 32 * 24;    // 16^3 x 48
  float* x3 = x2 + 16 * 16 * 16 * 48;    // 8^3  x 96
  float* x4 = x3 + 8 * 8 * 8 * 96;       // 4^3  x 192
  float* x5 = x4 + 4 * 4 * 4 * 192;      // 8^3  x 96
  float* x6 = x5 + 8 * 8 * 8 * 96;       // 16^3 x 48
  float* x7 = x6 + 16 * 16 * 16 * 48;    // 32^3 x 24
  float* x8 = x7 + 32 * 32 * 32 * 24;    // 64^3 x 24

  auto conv = [&](const float* in, const float* in2, const float* w,
                  const float* b, float* out, int iD, int iH, int iW, int Ci,
                  int oD, int oH, int oW, int Co, int stride, int padlo,
                  int dilate) {
    const int numVox = oD * oH * oW;
    dim3 grid((numVox + 63) / 64, (Co + 15) / 16, 1);
    switch (Ci) {
      case 24:
        conv3d_wmma_fast<24><<<grid, dim3(32), 0, stream>>>(
            in, in2, w, b, out, iD, iH, iW, oD, oH, oW, Co, stride, padlo, dilate);
        break;
      case 48:
        conv3d_wmma_fast<48><<<grid, dim3(32), 0, stream>>>(
            in, in2, w, b, out, iD, iH, iW, oD, oH, oW, Co, stride, padlo, dilate);
        break;
      case 96:
        conv3d_wmma_fast<96><<<grid, dim3(32), 0, stream>>>(
            in, in2, w, b, out, iD, iH, iW, oD, oH, oW, Co, stride, padlo, dilate);
        break;
      case 192:
        conv3d_wmma_fast<192><<<grid, dim3(32), 0, stream>>>(
            in, in2, w, b, out, iD, iH, iW, oD, oH, oW, Co, stride, padlo, dilate);
        break;
      default:  // Cin == 1 (first layer): generic path, no skip input needed.
        conv3d_wmma_generic<<<grid, dim3(32), 0, stream>>>(
            in, w, b, out, iD, iH, iW, Ci, oD, oH, oW, Co, stride, padlo);
        break;
    }
  };
  auto ln = [&](float* x, const float* s, const float* b, int C, int numVox) {
    layernorm_c<<<(numVox + 255) / 256, 256, 0, stream>>>(x, s, b, C, numVox);
  };

  // Front (attention collapses to constant pattern).
  front_colsum<<<1, 64, 0, stream>>>(P(N_OUT_W), P(N_LN2_B), S);
  fill_pattern<<<(64 * 64 * 64 + 255) / 256, 256, 0, stream>>>(S, x0);

  // Encoder: stride-2 SAME convs (pad_lo = 0).
  conv(x0, nullptr, P(N_CA_K), P(N_CA_B), x1, 64, 64, 64, 1,  32, 32, 32, 24,  2, 0, 0);
  conv(x1, nullptr, P(N_CB_K), P(N_CB_B), x2, 32, 32, 32, 24, 16, 16, 16, 48,  2, 0, 0);
  conv(x2, nullptr, P(N_CC_K), P(N_CC_B), x3, 16, 16, 16, 48,  8,  8,  8, 96,  2, 0, 0);
  conv(x3, nullptr, P(N_CD_K), P(N_CD_B), x4,  8,  8,  8, 96,  4,  4,  4, 192, 2, 0, 0);

  // Decoder: conv_transpose(k=3,s=2,'SAME') == stride-1 conv over 2x
  // zero-stuffed input with pad_lo=2 (implicit via dilate=1); skips fused.
  conv(x4, nullptr, P(N_DA_K), P(N_DA_B), x5,  4,  4,  4, 192, 8,  8,  8, 96,  1, 2, 1);
  ln(x5, P(N_DA_LN_S), P(N_DA_LN_B), 96, 8 * 8 * 8);
  conv(x5, x3, P(N_DB_K), P(N_DB_B), x6,  8,  8,  8, 96, 16, 16, 16, 48,  1, 2, 1);
  ln(x6, P(N_DB_LN_S), P(N_DB_LN_B), 48, 16 * 16 * 16);
  conv(x6, x2, P(N_DC_K), P(N_DC_B), x7, 16, 16, 16, 48, 32, 32, 32, 24,  1, 2, 1);
  ln(x7, P(N_DC_LN_S), P(N_DC_LN_B), 24, 32 * 32 * 32);
  conv(x7, x1, P(N_DD_K), P(N_DD_B), x8, 32, 32, 32, 24, 64, 64, 64, 24,  1, 2, 1);
  ln(x8, P(N_DD_LN_S), P(N_DD_LN_B), 24, 64 * 64 * 64);

  // Final stride-1 SAME conv (pad_lo = 1) to 1 channel, straight into d_out.
  conv(x8, nullptr, P(N_CO_K), P(N_CO_B), (float*)d_out,
       64, 64, 64, 24, 64, 64, 64, 1, 1, 1, 0);
}